// FusedTransformer_55173149884696
// MI455X (gfx1250) — compile-verified
//
#include <hip/hip_runtime.h>
#include <math.h>

#define B_   2
#define L_   2048
#define D_   1024
#define H_   16
#define DH_  64
#define NTOK (B_ * L_)

typedef __attribute__((ext_vector_type(2))) float v2f;
typedef __attribute__((ext_vector_type(8))) float v8f;

__device__ __forceinline__ v8f wmma4(v2f a, v2f b, v8f c) {
  // D = A(16x4,f32) x B(4x16,f32) + C(16x16,f32)
  return __builtin_amdgcn_wmma_f32_16x16x4_f32(
      /*neg_a=*/false, a, /*neg_b=*/false, b,
      /*c_mod=*/(short)0, c, /*reuse_a=*/false, /*reuse_b=*/false);
}

__device__ __forceinline__ float gelu_tanh(float x) {
  float x3 = x * x * x;
  return 0.5f * x * (1.0f + tanhf(0.7978845608028654f * (x + 0.044715f * x3)));
}

// ---------------------------------------------------------------- RMSNorm ---
__global__ void __launch_bounds__(256)
rmsnorm_kernel(const float* __restrict__ x, const float* __restrict__ scale,
               float* __restrict__ out) {
  const int row = blockIdx.x;
  const float* xr = x + (size_t)row * D_;
  float* orow = out + (size_t)row * D_;
  const int t = threadIdx.x;                 // 256 threads * 4 elems = 1024
  float4 v = ((const float4*)xr)[t];
  float ss = v.x * v.x + v.y * v.y + v.z * v.z + v.w * v.w;
  for (int off = 16; off >= 1; off >>= 1) ss += __shfl_xor(ss, off, 32);
  __shared__ float red[8];
  if ((t & 31) == 0) red[t >> 5] = ss;
  __syncthreads();
  float tot = 0.f;
#pragma unroll
  for (int i = 0; i < 8; ++i) tot += red[i];
  float r = rsqrtf(tot * (1.0f / (float)D_) + 1e-6f);
  float4 s4 = ((const float4*)scale)[t];
  float4 o;
  o.x = v.x * s4.x * r; o.y = v.y * s4.y * r;
  o.z = v.z * s4.z * r; o.w = v.w * s4.w * r;
  ((float4*)orow)[t] = o;
}

// ------------------------------------------------------------- GEMM (NT) ----
// C[M,N] = A[M,K] * W[N,K]^T ; flags: bit0 add R, bit1 gelu
// Block: 256 thr = 8 waves, tile 128x64; wave tile 16x64.
// Software-pipelined: next k-step's 5 fragments are loaded before the
// current step's 4 WMMAs issue, so loads stay in flight under the WMMAs.
__global__ void __launch_bounds__(256)
gemm_nt_kernel(const float* __restrict__ A, const float* __restrict__ W,
               const float* __restrict__ R, float* __restrict__ C,
               int M, int N, int K, int flags) {
  const int lane = threadIdx.x & 31;
  const int wave = threadIdx.x >> 5;  // wave row 0..7
  const int half = lane >> 4;
  const int l16 = lane & 15;
  const int rowBase = blockIdx.y * 128 + wave * 16;
  const int colBase = blockIdx.x * 64;

  const float* arow = A + (size_t)(rowBase + l16) * K + 2 * half;
  const float* b0r  = W + (size_t)(colBase + l16) * K + 2 * half;
  const float* b1r  = b0r + (size_t)16 * K;
  const float* b2r  = b0r + (size_t)32 * K;
  const float* b3r  = b0r + (size_t)48 * K;

  // one-shot prefetch of the row streams (lowered to global_prefetch_b8)
  __builtin_prefetch(arow, 0, 0);
  __builtin_prefetch(b0r, 0, 0);
  __builtin_prefetch(b1r, 0, 0);
  __builtin_prefetch(b2r, 0, 0);
  __builtin_prefetch(b3r, 0, 0);

  v8f acc0 = {0.f,0.f,0.f,0.f,0.f,0.f,0.f,0.f};
  v8f acc1 = acc0, acc2 = acc0, acc3 = acc0;

  v2f a_c  = *(const v2f*)(arow);
  v2f x0_c = *(const v2f*)(b0r);
  v2f x1_c = *(const v2f*)(b1r);
  v2f x2_c = *(const v2f*)(b2r);
  v2f x3_c = *(const v2f*)(b3r);

#pragma unroll 8
  for (int k = 0; k < K - 4; k += 4) {
    const int kn = k + 4;
    v2f a_n  = *(const v2f*)(arow + kn);
    v2f x0_n = *(const v2f*)(b0r + kn);
    v2f x1_n = *(const v2f*)(b1r + kn);
    v2f x2_n = *(const v2f*)(b2r + kn);
    v2f x3_n = *(const v2f*)(b3r + kn);
    acc0 = wmma4(a_c, x0_c, acc0);
    acc1 = wmma4(a_c, x1_c, acc1);
    acc2 = wmma4(a_c, x2_c, acc2);
    acc3 = wmma4(a_c, x3_c, acc3);
    a_c = a_n; x0_c = x0_n; x1_c = x1_n; x2_c = x2_n; x3_c = x3_n;
  }
  acc0 = wmma4(a_c, x0_c, acc0);
  acc1 = wmma4(a_c, x1_c, acc1);
  acc2 = wmma4(a_c, x2_c, acc2);
  acc3 = wmma4(a_c, x3_c, acc3);

#pragma unroll
  for (int r = 0; r < 8; ++r) {
    int row = rowBase + r + 8 * half;
    size_t i0 = (size_t)row * N + colBase + l16;
    float v0 = acc0[r], v1 = acc1[r], v2 = acc2[r], v3 = acc3[r];
    if (flags & 1) {
      v0 += R[i0]; v1 += R[i0 + 16]; v2 += R[i0 + 32]; v3 += R[i0 + 48];
    }
    if (flags & 2) {
      v0 = gelu_tanh(v0); v1 = gelu_tanh(v1);
      v2 = gelu_tanh(v2); v3 = gelu_tanh(v3);
    }
    C[i0] = v0; C[i0 + 16] = v1; C[i0 + 32] = v2; C[i0 + 48] = v3;
  }
}

// ------------------------------------------------------------------ RoPE ----
// In-place rotation of q,k inside qkv buffer. dr=5, d=25, pairs (i, i+25).
__global__ void __launch_bounds__(256)
rope_kernel(float* __restrict__ qkv, const float* __restrict__ pos,
            const float* __restrict__ pos_orig, const float* __restrict__ tm) {
  int idx = blockIdx.x * blockDim.x + threadIdx.x;
  if (idx >= NTOK * H_ * 25) return;
  int i   = idx % 25;
  int h   = (idx / 25) % H_;
  int tok = idx / (25 * H_);
  int c = i / 5, j = i % 5;
  int fi = j * H_ + h;                      // 0..79
  float theta;
  if (c < 4) {
    // f[fi] = exp(ln(pi) + fi*ln(10)/80)
    float freq = __expf(1.1447298858494002f + (float)fi * 0.028782313662425572f);
    int coord = (c < 2) ? 1 : 0;            // theta_h uses y, theta_w uses x
    int s = (c < 2) ? c : c - 2;            // 0 -> pos_orig, 1 -> pos
    const float* P = (s == 0) ? pos_orig : pos;
    float pv = P[(size_t)tok * 2 + coord] * 2.0f - 1.0f;
    theta = pv * freq;
  } else {
    // t_freqs[fi] = 100^(-fi/80) = exp(-fi*ln(100)/80)
    float tf = __expf(-(float)fi * 0.057564627324851145f);
    theta = tm[tok] * tf;
  }
  float cs = cosf(theta), sn = sinf(theta);
  float* qp = qkv + (size_t)tok * 3 * D_ + h * DH_;
  float* kp = qp + D_;
  float q1 = qp[i], q2 = qp[i + 25];
  qp[i] = q1 * cs - q2 * sn;  qp[i + 25] = q2 * cs + q1 * sn;
  float k1 = kp[i], k2 = kp[i + 25];
  kp[i] = k1 * cs - k2 * sn;  kp[i + 25] = k2 * cs + k1 * sn;
}

// ----------------------------------------------- cosine-sim q/k scaling ----
__global__ void __launch_bounds__(256)
cos_scale_kernel(float* __restrict__ qkv, const float* __restrict__ qk_scale) {
  const int lane = threadIdx.x & 31;
  const int wave = threadIdx.x >> 5;
  const int row = blockIdx.x * 8 + wave;    // (tok,h) row, 64 elems
  const int h = row % H_;
  const int tok = row / H_;
  float* qp = qkv + (size_t)tok * 3 * D_ + h * DH_;
  float* kp = qp + D_;
  float2 q2 = ((float2*)qp)[lane];
  float2 k2 = ((float2*)kp)[lane];
  float sq = q2.x * q2.x + q2.y * q2.y;
  float sk = k2.x * k2.x + k2.y * k2.y;
  for (int off = 16; off >= 1; off >>= 1) {
    sq += __shfl_xor(sq, off, 32);
    sk += __shfl_xor(sk, off, 32);
  }
  float s = sqrtf(qk_scale[h]);
  float qs = s * rsqrtf(sq + 1e-6f);
  float ks = s * rsqrtf(sk + 1e-6f);
  ((float2*)qp)[lane] = make_float2(q2.x * qs, q2.y * qs);
  ((float2*)kp)[lane] = make_float2(k2.x * ks, k2.y * ks);
}

// ------------------------------------------------------- Flash attention ----
// One wave per (16-query tile, b*h). qkv strided [tok, 3*D]; out [tok, H*DH].
__global__ void __launch_bounds__(32)
attn_kernel(const float* __restrict__ qkv, float* __restrict__ out) {
  const int lane = threadIdx.x & 31;
  const int half = lane >> 4, l16 = lane & 15;
  const int qt = blockIdx.x;                 // 0..127
  const int bh = blockIdx.y;                 // 0..31
  const int b = bh / H_, h = bh % H_;
  const size_t ts = 3 * D_;                  // token stride in floats

  const float* qrow = qkv + (size_t)(b * L_ + qt * 16 + l16) * ts + h * DH_ + 2 * half;
  v2f qf[16];
#pragma unroll
  for (int s = 0; s < 16; ++s) qf[s] = *(const v2f*)(qrow + 4 * s);

  float m[8], lsum[8];
#pragma unroll
  for (int r = 0; r < 8; ++r) { m[r] = -INFINITY; lsum[r] = 0.f; }
  v8f o0 = {0.f,0.f,0.f,0.f,0.f,0.f,0.f,0.f};
  v8f o1 = o0, o2 = o0, o3 = o0;

  __shared__ float P[16 * 17];               // stride 17: conflict-free
  const float* kbase = qkv + (size_t)(b * L_) * ts + D_ + h * DH_;
  const float* vbase = qkv + (size_t)(b * L_) * ts + 2 * D_ + h * DH_;

  for (int kt = 0; kt < L_ / 16; ++kt) {
    const float* krow = kbase + (size_t)(kt * 16 + l16) * ts + 2 * half;
    // two independent accumulation chains -> 2x WMMA overlap
    v8f S0 = {0.f,0.f,0.f,0.f,0.f,0.f,0.f,0.f};
    v8f S1 = S0;
#pragma unroll
    for (int s = 0; s < 16; s += 2) {
      v2f bf0 = *(const v2f*)(krow + 4 * s);
      v2f bf1 = *(const v2f*)(krow + 4 * s + 4);
      S0 = wmma4(qf[s], bf0, S0);
      S1 = wmma4(qf[s + 1], bf1, S1);
    }
    __syncthreads();                          // previous P reads complete
#pragma unroll
    for (int r = 0; r < 8; ++r) {
      float sv = S0[r] + S1[r];
      float mx = sv;
      for (int off = 8; off >= 1; off >>= 1) mx = fmaxf(mx, __shfl_xor(mx, off, 32));
      float mnew = fmaxf(m[r], mx);
      float alpha = __expf(m[r] - mnew);
      float p = __expf(sv - mnew);
      float rs = p;
      for (int off = 8; off >= 1; off >>= 1) rs += __shfl_xor(rs, off, 32);
      lsum[r] = lsum[r] * alpha + rs;
      m[r] = mnew;
      o0[r] *= alpha; o1[r] *= alpha; o2[r] *= alpha; o3[r] *= alpha;
      P[(r + 8 * half) * 17 + l16] = p;
    }
    __syncthreads();
    const float* vrow = vbase + (size_t)(kt * 16) * ts;
#pragma unroll
    for (int s = 0; s < 4; ++s) {
      int k0 = s * 4 + 2 * half;
      v2f pf;
      pf.x = P[l16 * 17 + k0];
      pf.y = P[l16 * 17 + k0 + 1];
      const float* v0p = vrow + (size_t)k0 * ts;
      const float* v1p = v0p + ts;
      v2f bv;
      bv.x = v0p[l16];       bv.y = v1p[l16];       o0 = wmma4(pf, bv, o0);
      bv.x = v0p[16 + l16];  bv.y = v1p[16 + l16];  o1 = wmma4(pf, bv, o1);
      bv.x = v0p[32 + l16];  bv.y = v1p[32 + l16];  o2 = wmma4(pf, bv, o2);
      bv.x = v0p[48 + l16];  bv.y = v1p[48 + l16];  o3 = wmma4(pf, bv, o3);
    }
  }
  float* obase = out + (size_t)(b * L_ + qt * 16) * D_ + h * DH_;
#pragma unroll
  for (int r = 0; r < 8; ++r) {
    float inv = 1.0f / lsum[r];
    float* op = obase + (size_t)(r + 8 * half) * D_;
    op[l16]      = o0[r] * inv;
    op[16 + l16] = o1[r] * inv;
    op[32 + l16] = o2[r] * inv;
    op[48 + l16] = o3[r] * inv;
  }
}

// ------------------------------------------------------------------ host ----
extern "C" void kernel_launch(void* const* d_in, const int* in_sizes, int n_in,
                              void* d_out, int out_size, void* d_ws, size_t ws_size,
                              hipStream_t stream) {
  (void)in_sizes; (void)n_in; (void)out_size; (void)ws_size;
  const float* x        = (const float*)d_in[0];
  const float* pos      = (const float*)d_in[1];
  const float* pos_orig = (const float*)d_in[2];
  const float* tm       = (const float*)d_in[3];
  const float* w_qkv    = (const float*)d_in[4];
  const float* w_out    = (const float*)d_in[5];
  const float* w_up     = (const float*)d_in[6];
  const float* w_down   = (const float*)d_in[7];
  const float* n1s      = (const float*)d_in[8];
  const float* n2s      = (const float*)d_in[9];
  const float* qksc     = (const float*)d_in[10];
  float* out = (float*)d_out;

  float* xn   = (float*)d_ws;                     // NTOK*D
  float* qkv  = xn  + (size_t)NTOK * D_;          // NTOK*3D
  float* attn = qkv + (size_t)NTOK * 3 * D_;      // NTOK*D
  float* up   = attn + (size_t)NTOK * D_;         // NTOK*4D

  // 1. xn = rmsnorm(x)
  rmsnorm_kernel<<<NTOK, 256, 0, stream>>>(x, n1s, xn);
  // 2. qkv = xn @ w_qkv^T        [4096 x 3072 x 1024]
  gemm_nt_kernel<<<dim3(3 * D_ / 64, NTOK / 128), 256, 0, stream>>>(
      xn, w_qkv, nullptr, qkv, NTOK, 3 * D_, D_, 0);
  // 3. RoPE on q,k in place
  rope_kernel<<<(NTOK * H_ * 25 + 255) / 256, 256, 0, stream>>>(qkv, pos, pos_orig, tm);
  // 4. cosine-sim normalize q,k
  cos_scale_kernel<<<NTOK * H_ / 8, 256, 0, stream>>>(qkv, qksc);
  // 5. attention -> attn [tok, H*DH]
  attn_kernel<<<dim3(L_ / 16, B_ * H_), 32, 0, stream>>>(qkv, attn);
  // 6. out = x + attn @ w_out^T  [4096 x 1024 x 1024]
  gemm_nt_kernel<<<dim3(D_ / 64, NTOK / 128), 256, 0, stream>>>(
      attn, w_out, x, out, NTOK, D_, D_, 1);
  // 7. xn = rmsnorm(out)
  rmsnorm_kernel<<<NTOK, 256, 0, stream>>>(out, n2s, xn);
  // 8. up = gelu(xn @ w_up^T)    [4096 x 4096 x 1024]
  gemm_nt_kernel<<<dim3(4 * D_ / 64, NTOK / 128), 256, 0, stream>>>(
      xn, w_up, nullptr, up, NTOK, 4 * D_, D_, 2);
  // 9. out += up @ w_down^T      [4096 x 1024 x 4096]
  gemm_nt_kernel<<<dim3(D_ / 64, NTOK / 128), 256, 0, stream>>>(
      up, w_down, out, out, NTOK, D_, 4 * D_, 1);
}